// NeuralODEHourly_90194313216230
// MI455X (gfx1250) — compile-verified
//
#include <hip/hip_runtime.h>
#include <math.h>

// ---------------------------------------------------------------------------
// NeuralODE hourly rollout, MI455X (gfx1250, wave32).
//   y_{t+1} = y_t + W2 . tanh( w1y * y_t + (W1u . u_t + b1) ) + b2
// Split: C[t] = W1u.u_t + b1 precomputed in parallel with V_WMMA_F32_16X16X4_F32
// (C = X[T,4] @ Wt[4,H], X[t] = [u0,u1,1,0], Wt rows = {W1[:,1],W1[:,2],b1,0}).
// Sequential part: single wave32, 8 hidden units/lane, HW tanh, shfl butterfly.
// ---------------------------------------------------------------------------

#define T_STEPS 65536
#define H_DIM   256

typedef __attribute__((ext_vector_type(2))) float v2f;
typedef __attribute__((ext_vector_type(8))) float v8f;

#if defined(__has_builtin)
#  if __has_builtin(__builtin_amdgcn_tanhf)
#    define FAST_TANH(x) __builtin_amdgcn_tanhf(x)
#  elif __has_builtin(__builtin_amdgcn_tanh_f32)
#    define FAST_TANH(x) __builtin_amdgcn_tanh_f32(x)
#  endif
#endif
#ifndef FAST_TANH
#  define FAST_TANH(x) tanhf(x)
#endif

// ---------------------------------------------------------------------------
// Kernel 1: parallel WMMA precompute of C[t][h] = W1[h][1]*u0 + W1[h][2]*u1 + b1[h]
// One wave per 16x16 (t,h) tile; 4 waves per block. 4096 x 16 = 65536 tiles.
//
// f32 WMMA operand layout assumption (ISA 7.12.2):
//   A (16x4, MxK): lane<16 -> M=lane, holds {K0,K1}; lane>=16 -> M=lane-16, {K2,K3}
//   B (4x16, KxN): lane<16 -> N=lane, holds {K0,K1}; lane>=16 -> N=lane-16, {K2,K3}
//   D (16x16)   : VGPR r -> M = r + (lane>=16 ? 8 : 0), N = lane & 15
// ---------------------------------------------------------------------------
__global__ __launch_bounds__(128)
void node_precompute_wmma(const float* __restrict__ U,
                          const float* __restrict__ W1,
                          const float* __restrict__ b1,
                          float* __restrict__ C)
{
    const int lane = threadIdx.x & 31;
    const int wave = threadIdx.x >> 5;
    const int tile = blockIdx.x * 4 + wave;          // [0, 65536)
    const int t0   = (tile >> 4) * 16;               // 16 h-tiles per t-tile row
    const int h0   = (tile & 15) * 16;
    const int half = lane >> 4;                      // 0: K={0,1}, 1: K={2,3}
    const int idx  = lane & 15;

    v2f a, b;
    if (half == 0) {
        // A rows = timesteps: X[t] = [u0, u1, 1, 0]  (K0,K1 here)
        a.x = U[(size_t)(t0 + idx) * 2 + 0];
        a.y = U[(size_t)(t0 + idx) * 2 + 1];
        // B cols = hidden units: Wt = {W1[:,1], W1[:,2], b1, 0}
        b.x = W1[(size_t)(h0 + idx) * 3 + 1];
        b.y = W1[(size_t)(h0 + idx) * 3 + 2];
    } else {
        a.x = 1.0f;              // K2: ones column selects b1 row of Wt
        a.y = 0.0f;              // K3: zero pad
        b.x = b1[h0 + idx];
        b.y = 0.0f;
    }

    v8f c = {};
    c = __builtin_amdgcn_wmma_f32_16x16x4_f32(
            /*neg_a=*/false, a, /*neg_b=*/false, b,
            /*c_mod=*/(short)0, c, /*reuse_a=*/false, /*reuse_b=*/false);

#pragma unroll
    for (int r = 0; r < 8; ++r) {
        const int m = r + half * 8;                  // D row (timestep offset)
        C[(size_t)(t0 + m) * H_DIM + h0 + idx] = c[r];
    }
}

// ---------------------------------------------------------------------------
// Kernel 2a: sequential rollout reading precomputed C. Exactly one wave32.
// Lane j owns hidden units [8j, 8j+8). C loads pipelined one step ahead.
// ---------------------------------------------------------------------------
__global__ __launch_bounds__(32)
void node_rollout_C(const float* __restrict__ C,
                    const float* __restrict__ W1,
                    const float* __restrict__ W2,
                    const float* __restrict__ b2,
                    const float* __restrict__ y0,
                    float* __restrict__ ys)
{
    const int lane = threadIdx.x;
    float w1y[8], w2v[8];
#pragma unroll
    for (int j = 0; j < 8; ++j) {
        const int h = lane * 8 + j;
        w1y[j] = W1[(size_t)h * 3 + 0];
        w2v[j] = W2[h];
    }
    const float bias2 = b2[0];
    float y = y0[0];

    const float4* __restrict__ Cv = (const float4*)C;   // 64 float4 per step
    const int base = lane * 2;

    float4 c0 = Cv[base + 0];
    float4 c1 = Cv[base + 1];

    for (int t = 0; t < T_STEPS; ++t) {
        // prefetch next step (clamped so no branch / no garbage regs)
        const int tn = (t + 1 < T_STEPS) ? (t + 1) : t;
        const float4 n0 = Cv[(size_t)tn * 64 + base + 0];
        const float4 n1 = Cv[(size_t)tn * 64 + base + 1];

        const float c[8] = { c0.x, c0.y, c0.z, c0.w, c1.x, c1.y, c1.z, c1.w };
        float s = 0.0f;
#pragma unroll
        for (int j = 0; j < 8; ++j) {
            const float z = fmaf(w1y[j], y, c[j]);
            s = fmaf(FAST_TANH(z), w2v[j], s);
        }
        // full butterfly: every lane ends with the complete dot product
#pragma unroll
        for (int off = 16; off >= 1; off >>= 1)
            s += __shfl_xor(s, off, 32);

        y = y + s + bias2;                 // DT = 1.0
        if (lane == 0) ys[t] = y;

        c0 = n0; c1 = n1;
    }
}

// ---------------------------------------------------------------------------
// Kernel 2b: fused fallback (no workspace): compute c on the fly from U.
// ---------------------------------------------------------------------------
__global__ __launch_bounds__(32)
void node_rollout_fused(const float* __restrict__ U,
                        const float* __restrict__ W1,
                        const float* __restrict__ b1,
                        const float* __restrict__ W2,
                        const float* __restrict__ b2,
                        const float* __restrict__ y0,
                        float* __restrict__ ys)
{
    const int lane = threadIdx.x;
    float w1y[8], wu0[8], wu1[8], bb[8], w2v[8];
#pragma unroll
    for (int j = 0; j < 8; ++j) {
        const int h = lane * 8 + j;
        w1y[j] = W1[(size_t)h * 3 + 0];
        wu0[j] = W1[(size_t)h * 3 + 1];
        wu1[j] = W1[(size_t)h * 3 + 2];
        bb[j]  = b1[h];
        w2v[j] = W2[h];
    }
    const float bias2 = b2[0];
    float y = y0[0];

    const float2* __restrict__ Uv = (const float2*)U;
    float2 u = Uv[0];

    for (int t = 0; t < T_STEPS; ++t) {
        const int tn = (t + 1 < T_STEPS) ? (t + 1) : t;
        const float2 un = Uv[tn];

        float s = 0.0f;
#pragma unroll
        for (int j = 0; j < 8; ++j) {
            float z = fmaf(wu1[j], u.y, bb[j]);
            z = fmaf(wu0[j], u.x, z);
            z = fmaf(w1y[j], y, z);
            s = fmaf(FAST_TANH(z), w2v[j], s);
        }
#pragma unroll
        for (int off = 16; off >= 1; off >>= 1)
            s += __shfl_xor(s, off, 32);

        y = y + s + bias2;
        if (lane == 0) ys[t] = y;
        u = un;
    }
}

// ---------------------------------------------------------------------------
extern "C" void kernel_launch(void* const* d_in, const int* in_sizes, int n_in,
                              void* d_out, int out_size, void* d_ws, size_t ws_size,
                              hipStream_t stream)
{
    const float* U  = (const float*)d_in[0];   // [T,2]
    const float* W1 = (const float*)d_in[1];   // [H,3]
    const float* b1 = (const float*)d_in[2];   // [H]
    const float* W2 = (const float*)d_in[3];   // [1,H]
    const float* b2 = (const float*)d_in[4];   // [1]
    const float* y0 = (const float*)d_in[5];   // [1,1]
    float* ys = (float*)d_out;                 // [T]

    const size_t needC = (size_t)T_STEPS * H_DIM * sizeof(float);  // 64 MB
    if (ws_size >= needC) {
        float* C = (float*)d_ws;
        // 65536 tiles / 4 waves per block
        node_precompute_wmma<<<(T_STEPS / 16) * (H_DIM / 16) / 4, 128, 0, stream>>>(
            U, W1, b1, C);
        node_rollout_C<<<1, 32, 0, stream>>>(C, W1, W2, b2, y0, ys);
    } else {
        node_rollout_fused<<<1, 32, 0, stream>>>(U, W1, b1, W2, b2, y0, ys);
    }
}